// QDDPM_cpu_24575802867828
// MI455X (gfx1250) — compile-verified
//
#include <hip/hip_runtime.h>

typedef __attribute__((ext_vector_type(2))) float v2f;
typedef __attribute__((ext_vector_type(8))) float v8f;

#define BATCH    256
#define N_STATE  65536                 // 2^16 amplitudes per batch item
#define TOTAL    (BATCH * N_STATE)     // 2^24 complex elements
#define ROWS     (TOTAL / 16)          // 2^20 rows of 16
#define TILES    (ROWS / 16)           // 65536 WMMA tiles per pass

// ---------------------------------------------------------------------------
// Split interleaved complex input into planar Re / Im (better coalescing, and
// WMMA operand gathers become plain float loads).
// ---------------------------------------------------------------------------
__global__ __launch_bounds__(256) void split_kernel(const float* __restrict__ in,
                                                    float* __restrict__ Sr,
                                                    float* __restrict__ Si) {
    unsigned i = blockIdx.x * 256u + threadIdx.x;
    Sr[i] = in[2u * i];
    Si[i] = in[2u * i + 1u];
}

// ---------------------------------------------------------------------------
// Apply U4 = U (x) U (x) U (x) U (16x16 complex) along the nibble at bit
// position SHIFT of the state index.  One wave owns one 16-row x 16-col tile:
//   Y = X * W,  W = U4^T  (complex)  ->  4 real GEMM terms, f32 WMMA 16x16x4.
// In-place safe: each tile exclusively owns its 16x16 elements.
// SIGN: fold the CZ diagonal (all adjacent-qubit pairs) into the final pass.
// ---------------------------------------------------------------------------
template <int SHIFT, int SIGN>
__global__ __launch_bounds__(256) void gate4_pass(float* __restrict__ Sr,
                                                  float* __restrict__ Si) {
    __shared__ float Wr[16][16], Wi[16][16], Wn[16][16];

    // Build W[k][j] = U4[j,k] = prod_b U[j_b, k_b]   (256 entries, 1/thread)
    {
        const float tx = 0.15f, ty = 0.05f;            // RX(0.3), RY(0.1) half-angles
        const float cx = cosf(tx), sx = sinf(tx);
        const float cy = cosf(ty), sy = sinf(ty);
        // U = RY @ RX  (complex 2x2)
        const float ur[2][2] = {{cy * cx, -sy * cx}, {sy * cx,  cy * cx}};
        const float ui[2][2] = {{sy * sx, -cy * sx}, {-cy * sx, -sy * sx}};
        int t = threadIdx.x;
        int k = t >> 4, j = t & 15;
        float pr = 1.0f, pi = 0.0f;
        #pragma unroll
        for (int b = 0; b < 4; ++b) {
            int jb = (j >> b) & 1, kb = (k >> b) & 1;
            float qr = ur[jb][kb], qi = ui[jb][kb];
            float nr = pr * qr - pi * qi;
            float ni = pr * qi + pi * qr;
            pr = nr; pi = ni;
        }
        Wr[k][j] = pr; Wi[k][j] = pi; Wn[k][j] = -pi;
    }
    __syncthreads();

    const unsigned lane = threadIdx.x & 31u;
    const unsigned half = lane >> 4;        // 0: K pair {0,1}, 1: K pair {2,3}
    const unsigned lm   = lane & 15u;       // A: row;  B/D: column

    // B operands (wave-invariant): chunk c covers K = 4c .. 4c+3
    v2f br[4], bi[4], bn[4];
    #pragma unroll
    for (int c = 0; c < 4; ++c) {
        unsigned k = 4u * c + 2u * half;
        br[c].x = Wr[k][lm];  br[c].y = Wr[k + 1][lm];
        bi[c].x = Wi[k][lm];  bi[c].y = Wi[k + 1][lm];
        bn[c].x = Wn[k][lm];  bn[c].y = Wn[k + 1][lm];
    }

    const unsigned s = 1u << SHIFT;
    const unsigned lowmask = s - 1u;
    const unsigned waveId = (blockIdx.x * blockDim.x + threadIdx.x) >> 5;
    const unsigned nWaves = (gridDim.x * blockDim.x) >> 5;

    for (unsigned tile = waveId; tile < TILES; tile += nWaves) {
        // A operands: lane holds row lm, K-elements {k, k+1} per chunk
        const unsigned rRow = tile * 16u + lm;
        const unsigned base = ((rRow >> SHIFT) << (SHIFT + 4)) | (rRow & lowmask);
        v2f ar[4], ai[4];
        #pragma unroll
        for (int c = 0; c < 4; ++c) {
            unsigned k  = 4u * c + 2u * half;
            unsigned a0 = base + k * s;
            ar[c].x = Sr[a0];  ar[c].y = Sr[a0 + s];
            ai[c].x = Si[a0];  ai[c].y = Si[a0 + s];
        }

        v8f accR = {}; v8f accI = {};
        #pragma unroll
        for (int c = 0; c < 4; ++c) {
            // Yr = Xr*Wr + Xi*(-Wi) ; Yi = Xr*Wi + Xi*Wr
            accR = __builtin_amdgcn_wmma_f32_16x16x4_f32(false, ar[c], false, br[c],
                                                         (short)0, accR, false, false);
            accR = __builtin_amdgcn_wmma_f32_16x16x4_f32(false, ai[c], false, bn[c],
                                                         (short)0, accR, false, false);
            accI = __builtin_amdgcn_wmma_f32_16x16x4_f32(false, ar[c], false, bi[c],
                                                         (short)0, accI, false, false);
            accI = __builtin_amdgcn_wmma_f32_16x16x4_f32(false, ai[c], false, br[c],
                                                         (short)0, accI, false, false);
        }

        // D layout: lane<16 -> M = vgpr, lane>=16 -> M = vgpr+8; N = lm
        #pragma unroll
        for (int j = 0; j < 8; ++j) {
            unsigned m    = (unsigned)j + 8u * half;
            unsigned rOut = tile * 16u + m;
            unsigned bOut = ((rOut >> SHIFT) << (SHIFT + 4)) | (rOut & lowmask);
            unsigned addr = bOut + lm * s;
            float yr = accR[j], yi = accI[j];
            if (SIGN) {
                unsigned v = addr & 0xFFFFu;                    // index within state
                unsigned pb = v & (v >> 1) & 0x7FFFu;           // all adjacent pairs
                if (__popc(pb) & 1) { yr = -yr; yi = -yi; }
            }
            Sr[addr] = yr;
            Si[addr] = yi;
        }
    }
}

// ---------------------------------------------------------------------------
// Ancilla probabilities: one block per (batch, ancilla-outcome) pair.
// ---------------------------------------------------------------------------
__global__ __launch_bounds__(256) void probs_kernel(const float* __restrict__ Sr,
                                                    const float* __restrict__ Si,
                                                    float* __restrict__ p) {
    unsigned b = blockIdx.x >> 4, a = blockIdx.x & 15u;
    unsigned base = b * (unsigned)N_STATE + a * 4096u;
    float sum = 0.0f;
    for (unsigned d = threadIdx.x; d < 4096u; d += 256u) {
        float r = Sr[base + d], im = Si[base + d];
        sum += r * r + im * im;
    }
    __shared__ float red[256];
    red[threadIdx.x] = sum;
    __syncthreads();
    for (unsigned st = 128u; st > 0u; st >>= 1) {
        if (threadIdx.x < st) red[threadIdx.x] += red[threadIdx.x + st];
        __syncthreads();
    }
    if (threadIdx.x == 0) p[blockIdx.x] = red[0];
}

// Deterministic categorical sample per batch item (splitmix64 hash -> inverse CDF).
__global__ void select_kernel(const float* __restrict__ p,
                              int* __restrict__ m,
                              float* __restrict__ inv_norm) {
    int b = threadIdx.x;   // launched with 256 threads
    unsigned long long z = (unsigned long long)(b + 1) * 0x9E3779B97F4A7C15ull
                         + 42ull * 0xBF58476D1CE4E5B9ull;
    z ^= z >> 30; z *= 0xBF58476D1CE4E5B9ull;
    z ^= z >> 27; z *= 0x94D049BB133111EBull;
    z ^= z >> 31;
    float u = (float)((z >> 40) & 0xFFFFFFull) * (1.0f / 16777216.0f);
    const float* pb = p + b * 16;
    float tot = 0.0f;
    for (int a = 0; a < 16; ++a) tot += pb[a];
    float thr = u * tot, c = 0.0f;
    int sel = 15;
    for (int a = 0; a < 16; ++a) {
        c += pb[a];
        if (c >= thr) { sel = a; break; }
    }
    m[b] = sel;
    inv_norm[b] = rsqrtf(pb[sel]);
}

// Gather selected block, normalize, re-interleave to [B, 4096, 2] f32.
__global__ __launch_bounds__(256) void output_kernel(const float* __restrict__ Sr,
                                                     const float* __restrict__ Si,
                                                     const int* __restrict__ m,
                                                     const float* __restrict__ inv_norm,
                                                     float* __restrict__ out) {
    unsigned i = blockIdx.x * 256u + threadIdx.x;   // 0 .. B*4096
    unsigned b = i >> 12, d = i & 4095u;
    unsigned idx = b * (unsigned)N_STATE + (unsigned)m[b] * 4096u + d;
    float s = inv_norm[b];
    out[2u * i]     = Sr[idx] * s;
    out[2u * i + 1] = Si[idx] * s;
}

// ---------------------------------------------------------------------------
extern "C" void kernel_launch(void* const* d_in, const int* in_sizes, int n_in,
                              void* d_out, int out_size, void* d_ws, size_t ws_size,
                              hipStream_t stream) {
    const float* in = (const float*)d_in[0];       // [256, 65536, 2] f32
    float* out = (float*)d_out;                    // [256, 4096, 2] f32

    float* Sr = (float*)d_ws;                      // 2^24 floats (64 MB)
    float* Si = Sr + TOTAL;                        // 2^24 floats (64 MB)
    float* p  = Si + TOTAL;                        // 256*16
    float* nv = p + BATCH * 16;                    // 256
    int*   m  = (int*)(nv + BATCH);                // 256

    split_kernel<<<TOTAL / 256, 256, 0, stream>>>(in, Sr, Si);

    const int gateBlocks = 1024;                   // 8192 waves, 8 tiles each
    for (int layer = 0; layer < 6; ++layer) {
        gate4_pass<12, 0><<<gateBlocks, 256, 0, stream>>>(Sr, Si);
        gate4_pass< 8, 0><<<gateBlocks, 256, 0, stream>>>(Sr, Si);
        gate4_pass< 4, 0><<<gateBlocks, 256, 0, stream>>>(Sr, Si);
        gate4_pass< 0, 1><<<gateBlocks, 256, 0, stream>>>(Sr, Si);  // fold CZ sign
    }

    probs_kernel<<<BATCH * 16, 256, 0, stream>>>(Sr, Si, p);
    select_kernel<<<1, 256, 0, stream>>>(p, m, nv);
    output_kernel<<<(BATCH * 4096) / 256, 256, 0, stream>>>(Sr, Si, m, nv, out);
}